// GENConv_57604101374215
// MI455X (gfx1250) — compile-verified
//
#include <hip/hip_runtime.h>
#include <hip/hip_bf16.h>
#include <math.h>

#define EPSV 1e-7f

typedef __attribute__((ext_vector_type(2))) float v2f;
typedef __attribute__((ext_vector_type(4))) float v4f;
typedef __attribute__((ext_vector_type(8))) float v8f;

// float atomic max via signed-max / unsigned-min bit trick (works for mixed signs,
// init value -inf). Lowers to native global_atomic_max_i32 / global_atomic_min_u32.
__device__ __forceinline__ void atomicMaxF32(float* addr, float val) {
    if (val >= 0.0f) {
        atomicMax((int*)addr, __float_as_int(val));
    } else {
        atomicMin((unsigned int*)addr, __float_as_uint(val));
    }
}

// ---------------------------------------------------------------------------
// Kernel 1: init workspace: max=-inf, den=0, num=0  (vectorized float4)
// ---------------------------------------------------------------------------
__global__ void genconv_init_kernel(float* __restrict__ maxb,
                                    float* __restrict__ den,
                                    float* __restrict__ num,
                                    int n4) {
    int i = blockIdx.x * blockDim.x + threadIdx.x;
    if (i >= n4) return;
    float ninf = -__builtin_inff();
    v4f vninf = {ninf, ninf, ninf, ninf};
    v4f vz    = {0.f, 0.f, 0.f, 0.f};
    ((v4f*)maxb)[i] = vninf;
    ((v4f*)den)[i]  = vz;
    ((v4f*)num)[i]  = vz;
}

// ---------------------------------------------------------------------------
// Kernel 2: per-edge segment max of beta*msg. One wave (32 lanes) per edge,
// 2 channels per lane -> coalesced 256B row load per edge.
// ---------------------------------------------------------------------------
__global__ void genconv_edge_max_kernel(const float* __restrict__ x,
                                        const long long* __restrict__ ei,
                                        const float* __restrict__ beta_p,
                                        float* __restrict__ maxb,
                                        int E) {
    int wid = (blockIdx.x * blockDim.x + threadIdx.x) >> 5;
    if (wid >= E) return;
    int lane = threadIdx.x & 31;
    int dst = (int)ei[wid];
    int src = (int)ei[(long)E + wid];
    float beta = beta_p[0];

    v2f xv = *(const v2f*)&x[(long)src * 64 + lane * 2];
    float m0 = fmaxf(xv.x, 0.0f) + EPSV;
    float m1 = fmaxf(xv.y, 0.0f) + EPSV;

    float* p = maxb + (long)dst * 64 + lane * 2;
    atomicMaxF32(p,     beta * m0);
    atomicMaxF32(p + 1, beta * m1);
}

// ---------------------------------------------------------------------------
// Kernel 3: e = exp(beta*msg - max);  den += e;  num += msg*e
// ---------------------------------------------------------------------------
__global__ void genconv_edge_sum_kernel(const float* __restrict__ x,
                                        const long long* __restrict__ ei,
                                        const float* __restrict__ beta_p,
                                        const float* __restrict__ maxb,
                                        float* __restrict__ den,
                                        float* __restrict__ num,
                                        int E) {
    int wid = (blockIdx.x * blockDim.x + threadIdx.x) >> 5;
    if (wid >= E) return;
    int lane = threadIdx.x & 31;
    int dst = (int)ei[wid];
    int src = (int)ei[(long)E + wid];
    float beta = beta_p[0];

    v2f xv = *(const v2f*)&x[(long)src * 64 + lane * 2];
    float m0 = fmaxf(xv.x, 0.0f) + EPSV;
    float m1 = fmaxf(xv.y, 0.0f) + EPSV;

    long off = (long)dst * 64 + lane * 2;
    v2f mm = *(const v2f*)&maxb[off];           // finite: this edge contributed in pass 2
    float e0 = __expf(beta * m0 - mm.x);
    float e1 = __expf(beta * m1 - mm.y);

    atomicAdd(den + off,     e0);
    atomicAdd(den + off + 1, e1);
    atomicAdd(num + off,     m0 * e0);
    atomicAdd(num + off + 1, m1 * e1);
}

// ---------------------------------------------------------------------------
// Kernel 4: out = (num/(den+1e-16)) @ W^T + b   via v_wmma_f32_16x16x4_f32.
// Block = 128 threads = 4 waves. Block owns 16 node rows; wave w owns output
// cols [16w, 16w+16). A tile (16x64 agg) staged once in LDS.
//
// f32 WMMA fragment layouts (ISA 7.12.2):
//   A 16x4 : lanes 0-15 -> M=lane, VGPR0=K+0,VGPR1=K+1 ; lanes 16-31 -> K+2,K+3
//   B 4x16 : VGPR0: half0->K+0, half1->K+2 ; VGPR1: half0->K+1, half1->K+3 ; N=lane%16
//   C/D    : VGPR v: lanes 0-15 -> M=v, lanes 16-31 -> M=v+8 ; N=lane%16
// ---------------------------------------------------------------------------
__global__ void genconv_gemm_wmma_kernel(const float* __restrict__ num,
                                         const float* __restrict__ den,
                                         const float* __restrict__ W,
                                         const float* __restrict__ bias,
                                         float* __restrict__ out,
                                         int N) {
    __shared__ float lds_agg[16 * 64];

    int tid = threadIdx.x;
    int m0  = blockIdx.x * 16;
    long base = (long)m0 * 64;

#pragma unroll
    for (int i = 0; i < 8; ++i) {
        int idx = tid + i * 128;
        float a = 0.0f;
        if (m0 + (idx >> 6) < N)
            a = num[base + idx] / (den[base + idx] + 1e-16f);
        lds_agg[idx] = a;
    }
    __syncthreads();

    int wv   = tid >> 5;
    int lane = tid & 31;
    int half = lane >> 4;   // 0: lanes 0-15, 1: lanes 16-31
    int lm   = lane & 15;
    int n0   = wv * 16;

    v8f acc = {};
#pragma unroll
    for (int k = 0; k < 64; k += 4) {
        int kk = k + half * 2;
        v2f afrag = *(const v2f*)&lds_agg[lm * 64 + kk];          // A[M=lm][kk,kk+1]
        v2f bfrag = *(const v2f*)&W[(n0 + lm) * 64 + kk];         // B[kk,kk+1][N=lm] = W[n][k]
        acc = __builtin_amdgcn_wmma_f32_16x16x4_f32(
            false, afrag, false, bfrag, (short)0, acc, false, false);
    }

    float bb = bias[n0 + lm];
#pragma unroll
    for (int v = 0; v < 8; ++v) {
        int row = m0 + v + half * 8;
        if (row < N)
            out[(long)row * 64 + n0 + lm] = acc[v] + bb;
    }
}

// ---------------------------------------------------------------------------
extern "C" void kernel_launch(void* const* d_in, const int* in_sizes, int n_in,
                              void* d_out, int out_size, void* d_ws, size_t ws_size,
                              hipStream_t stream) {
    const float*     x     = (const float*)d_in[0];       // [N,64]
    const long long* ei    = (const long long*)d_in[1];   // [2,E] int64: row0=dst, row1=src
    const float*     W     = (const float*)d_in[2];       // [64,64]
    const float*     bias  = (const float*)d_in[3];       // [64]
    const float*     beta  = (const float*)d_in[4];       // [1]
    float*           out   = (float*)d_out;

    int N = in_sizes[0] / 64;
    int E = in_sizes[1] / 2;

    float* maxb = (float*)d_ws;                 // N*64 floats
    float* den  = maxb + (size_t)N * 64;        // N*64 floats
    float* num  = den  + (size_t)N * 64;        // N*64 floats

    int n4 = N * 16;                            // float4 count per buffer
    genconv_init_kernel<<<(n4 + 255) / 256, 256, 0, stream>>>(maxb, den, num, n4);

    int edge_blocks = (E + 7) / 8;              // 8 waves (edges) per 256-thread block
    genconv_edge_max_kernel<<<edge_blocks, 256, 0, stream>>>(x, ei, beta, maxb, E);
    genconv_edge_sum_kernel<<<edge_blocks, 256, 0, stream>>>(x, ei, beta, maxb, den, num, E);

    int gemm_blocks = (N + 15) / 16;
    genconv_gemm_wmma_kernel<<<gemm_blocks, 128, 0, stream>>>(num, den, W, bias, out, N);
}